// PeriodicRadiusGraph_47519518163698
// MI455X (gfx1250) — compile-verified
//
#include <hip/hip_runtime.h>
#include <hip/hip_bf16.h>

// ---------------------------------------------------------------------------
// PeriodicRadiusGraph for MI455X (gfx1250).
// Output [1024,1024,27,3] f32 = 339.7 MB  -> store-bandwidth bound (~15 us at
// 23.3 TB/s). Kernel 2 streams the output with fully coalesced float4 stores.
// Kernel 1 computes cart = frac @ cell on the matrix unit (v_wmma_f32_16x16x4_f32)
// plus the 27 image-shift vectors, into the workspace.
// ---------------------------------------------------------------------------

typedef float v2f __attribute__((ext_vector_type(2)));
typedef float v8f __attribute__((ext_vector_type(8)));

#define N_ATOMS   1024
#define N_SHIFT   27
#define CUT2      25.0f          // 5.0^2
#define ZERO_SHIFT 13u           // (a,b,c)=(1,1,1) in {-1,0,1}^3, index a*9+b*3+c
// workspace layout (floats): [0,4096) cart padded float4; [4096,4224) shifts
#define WS_SHIFT_OFF 4096

// ---- Kernel 1: cart = frac @ cell via V_WMMA_F32_16X16X4_F32 ---------------
// One wave per 16-atom tile. A = 16x4 (frac rows, K=3 used), B = 4x16 (cell,
// N=3 used), D = 16x16 f32, of which columns 0..2 are the cartesian coords.
// ISA layout (32-bit, wave32):
//   A: VGPR0 = {lanes0-15: K=0, lanes16-31: K=2}; VGPR1 = {K=1, K=3}
//   B: VGPR0 = {lanes0-15: row K=0 (N=lane), lanes16-31: row K=2}; VGPR1 = {K=1, K=3}
//   D: VGPR r, lanes0-15 -> (M=r, N=lane); lanes16-31 -> (M=r+8, N=lane-16)
__global__ __launch_bounds__(32)
void prg_cart_wmma_kernel(const float* __restrict__ frac,
                          const float* __restrict__ cell,
                          float* __restrict__ ws) {
    const int lane = threadIdx.x;          // 0..31
    const int tile = blockIdx.x;           // 0..63
    const bool hi  = (lane >= 16);
    const int  n   = lane & 15;            // N (and M within half) index
    const int  row = tile * 16 + n;        // frac row for A operand

    // A operand: frac[row][K]
    v2f a;
    a.x = hi ? frac[row * 3 + 2] : frac[row * 3 + 0];   // K=2 : K=0
    a.y = hi ? 0.0f              : frac[row * 3 + 1];   // K=3 : K=1

    // B operand: cell[K][n] (n<3), padded with zeros
    v2f b;
    b.x = (n < 3) ? (hi ? cell[2 * 3 + n] : cell[0 * 3 + n]) : 0.0f;  // K=2 : K=0
    b.y = (n < 3) ? (hi ? 0.0f            : cell[1 * 3 + n]) : 0.0f;  // K=3 : K=1

    v8f c = {};
    c = __builtin_amdgcn_wmma_f32_16x16x4_f32(
            /*neg_a=*/false, a, /*neg_b=*/false, b,
            /*c_mod=*/(short)0, c, /*reuse_a=*/false, /*reuse_b=*/false);

    // Scatter D columns 0..2 into padded cart[atom][4]; column 3 gets zeros.
    const int mbase = hi ? 8 : 0;
    if (n < 3) {
        #pragma unroll
        for (int r = 0; r < 8; ++r)
            ws[(tile * 16 + mbase + r) * 4 + n] = c[r];
    } else if (n == 3) {
        #pragma unroll
        for (int r = 0; r < 8; ++r)
            ws[(tile * 16 + mbase + r) * 4 + 3] = 0.0f;
    }

    // shift_cart = S @ cell for the 27 image shifts (block 0 only)
    if (tile == 0 && lane < N_SHIFT) {
        const float sx = (float)(lane / 9)       - 1.0f;
        const float sy = (float)((lane / 3) % 3) - 1.0f;
        const float sz = (float)(lane % 3)       - 1.0f;
        #pragma unroll
        for (int k = 0; k < 3; ++k)
            ws[WS_SHIFT_OFF + lane * 4 + k] =
                sx * cell[0 * 3 + k] + sy * cell[1 * 3 + k] + sz * cell[2 * 3 + k];
        ws[WS_SHIFT_OFF + lane * 4 + 3] = 0.0f;
    }
}

// ---- Kernel 2: stream masked displacements, float4 per thread --------------
// Quad g covers output floats [4g, 4g+4) == tail of triple t0 = 4g/3 and head
// of triple t0+1. Each thread computes both triples' masked displacement and
// assembles one aligned float4 -> every wave store is 512 contiguous bytes.
#define QPT 32u   // quads per thread; block emits 256*32*16B = 128 KB

__global__ __launch_bounds__(256)
void prg_disp_kernel(const float* __restrict__ ws, float* __restrict__ out) {
    __shared__ float4 s_cart[N_ATOMS];
    __shared__ float4 s_shc[N_SHIFT];

    const float4* wsc = (const float4*)ws;
    #pragma unroll
    for (int idx = threadIdx.x; idx < N_ATOMS; idx += 256)
        s_cart[idx] = wsc[idx];
    if (threadIdx.x < N_SHIFT)
        s_shc[threadIdx.x] = wsc[N_ATOMS + threadIdx.x];
    __syncthreads();

    float4* __restrict__ out4 = (float4*)out;
    const unsigned base = blockIdx.x * (256u * QPT) + threadIdx.x;

    for (unsigned q = 0; q < QPT; ++q) {
        const unsigned g  = base + q * 256u;       // quad index
        const unsigned f0 = 4u * g;                // first float index
        const unsigned t0 = f0 / 3u;               // first triple
        const unsigned r  = f0 - 3u * t0;          // rotation 0..2

        float a[6];
        #pragma unroll
        for (int k = 0; k < 2; ++k) {
            const unsigned T  = t0 + (unsigned)k;
            const unsigned s  = T % 27u;
            const unsigned ij = T / 27u;
            const unsigned j  = ij & 1023u;
            const unsigned i  = ij >> 10;
            const float4 cj = s_cart[j];
            const float4 ci = s_cart[i];
            const float4 sc = s_shc[s];
            const float dx = cj.x + sc.x - ci.x;
            const float dy = cj.y + sc.y - ci.y;
            const float dz = cj.z + sc.z - ci.z;
            const float d2 = dx * dx + dy * dy + dz * dz;
            const bool keep = (d2 < CUT2) && !((i == j) && (s == ZERO_SHIFT));
            const float m = keep ? 1.0f : 0.0f;
            a[k * 3 + 0] = dx * m;
            a[k * 3 + 1] = dy * m;
            a[k * 3 + 2] = dz * m;
        }

        float4 o;   // rotate by r without indexed (scratch) access
        o.x = (r == 0u) ? a[0] : ((r == 1u) ? a[1] : a[2]);
        o.y = (r == 0u) ? a[1] : ((r == 1u) ? a[2] : a[3]);
        o.z = (r == 0u) ? a[2] : ((r == 1u) ? a[3] : a[4]);
        o.w = (r == 0u) ? a[3] : ((r == 1u) ? a[4] : a[5]);
        out4[g] = o;
    }
}

extern "C" void kernel_launch(void* const* d_in, const int* in_sizes, int n_in,
                              void* d_out, int out_size, void* d_ws, size_t ws_size,
                              hipStream_t stream) {
    const float* frac = (const float*)d_in[0];   // [1024,3] f32
    const float* cell = (const float*)d_in[1];   // [3,3]   f32
    float* out = (float*)d_out;                  // [1024,1024,27,3] f32
    float* ws  = (float*)d_ws;                   // >= 4224 floats used

    // Kernel 1: cart + shift_cart into workspace (64 single-wave tiles)
    prg_cart_wmma_kernel<<<N_ATOMS / 16, 32, 0, stream>>>(frac, cell, ws);

    // Kernel 2: stream output. quads = out_size/4, threads/block=256, QPT quads each.
    const unsigned quads  = (unsigned)(out_size / 4);            // 21,233,664
    const unsigned blocks = quads / (256u * QPT);                // 2592
    prg_disp_kernel<<<blocks, 256, 0, stream>>>(ws, out);
}